// EPG_37572373906034
// MI455X (gfx1250) — compile-verified
//
#include <hip/hip_runtime.h>
#include <math.h>

// ---------------------------------------------------------------------------
// EPG-GRE F0 simulation for MI455X (gfx1250).
//
// Design: compute-bound (~2.4G fp32 FMA), memory tiny -> register-resident
// state, one voxel per lane (wave32), K truncated to 32 (output-exact),
// active-band guards k <= min(p, 63-p, 31) (output-exact).
// Rotation coefficient table staged to LDS via the CDNA5 Tensor Data Mover
// (tensor_load_to_lds + s_wait_tensorcnt), then broadcast to SGPRs with
// v_readfirstlane so the inner FMAs use scalar coefficient operands.
//
// Round-4 result: amdgpu_num_vgpr(256) forced heavy scratch spilling
// (vmem 12 -> 1034). Reverted: the natural ~360-VGPR allocation with
// s_set_vgpr_msb is strictly better (no spills; SALU co-issues).
// Kept: Z relaxation folded into rotation write-back (Z never shifts).
// ---------------------------------------------------------------------------

#define NP2  64          // number of RF pulses
#define NV   32768       // number of voxels
#define KS   32          // truncated EPG order (exact for F0, see analysis)
#define ASTR 16          // floats per pulse in the coefficient table

typedef __attribute__((ext_vector_type(4))) unsigned int uint32x4;
typedef __attribute__((ext_vector_type(8))) int          int32x8;
typedef __attribute__((ext_vector_type(4))) int          int32x4;

__device__ __forceinline__ float rfl(float x) {
    union { float f; int i; } u;
    u.f = x;
    u.i = __builtin_amdgcn_readfirstlane(u.i);   // -> SGPR (wave-uniform value)
    return u.f;
}

// ---------------------------------------------------------------------------
// Main EPG recurrence. One voxel per lane, state fully in VGPRs.
// ---------------------------------------------------------------------------
__global__ __launch_bounds__(256)
void epg_main(
    const float* __restrict__ Atab,    // [NP2][ASTR] floats in d_ws
    const float* __restrict__ qmaps,   // [3][NV]: PD, T1, T2
    const float* __restrict__ TRp,     // [1]
    float* __restrict__ out)           // [NV][NP2]
{
    __shared__ __align__(16) float As[NP2 * ASTR];     // 4 KB coefficient table

    // ---- Stage the coefficient table into LDS with the Tensor Data Mover.
    // D# built per CDNA5 ISA ch.8: group0 = {flags, lds_addr, gaddr_lo,
    // gaddr_hi|type=2}, group1 carries data_size/dims/tile/stride.
    if (threadIdx.x < 32) {            // one wave issues; TDM ignores EXEC
        // Low 32 bits of a generic LDS pointer are the wave-relative LDS
        // byte offset (the LDS aperture uses them directly as the address).
        unsigned lds_off = (unsigned)(unsigned long long)(void*)&As[0];
        unsigned long long ga = (unsigned long long)Atab;
        const unsigned nelem = NP2 * ASTR;             // 1024 dwords = 4 KB

        uint32x4 g0;
        g0[0] = 1u;                                     // count=1, user mode
        g0[1] = lds_off;                                // lds_addr (bytes)
        g0[2] = (unsigned)(ga & 0xffffffffu);           // global_addr[31:0]
        g0[3] = (unsigned)((ga >> 32) & 0x01ffffffu)    // global_addr[56:32]
              | (2u << 30);                             // type=2 ("image")

        int32x8 g1;
        g1[0] = (int)(2u << 16);                        // data_size=2 (4B), no mask
        g1[1] = (int)((nelem & 0xffffu) << 16);         // tensor_dim0[15:0]
        g1[2] = (int)((nelem >> 16) | (1u << 16));      // tensor_dim0 hi | tensor_dim1=1
        g1[3] = (int)(nelem << 16);                     // tile_dim0 = 1024
        g1[4] = 1;                                      // tile_dim1 = 1
        g1[5] = (int)nelem;                             // tensor_dim0_stride lo
        g1[6] = 0;                                      // stride0 hi, stride1 lo
        g1[7] = 0;                                      // stride1 hi

        int32x4 gz4 = {0, 0, 0, 0};                     // groups 2/3 unused (<=2D)
        int32x8 gz8 = {0, 0, 0, 0, 0, 0, 0, 0};        // unused trailing group
        // clang-23 / therock toolchain: 6-arg form
        __builtin_amdgcn_tensor_load_to_lds(g0, g1, gz4, gz4, gz8, 0);
        __builtin_amdgcn_s_wait_tensorcnt(0);           // TENSORcnt == 0
    }
    __syncthreads();

    // ---- Per-voxel setup
    const int v  = blockIdx.x * blockDim.x + threadIdx.x;
    const float PD = qmaps[v];
    const float T1 = qmaps[NV + v];
    const float T2 = qmaps[2 * NV + v];
    const float TR = TRp[0];
    const float E1 = expf(-TR / T1);
    const float E2 = expf(-TR / T2);
    const float B0 = 1.0f - E1;

    // Register-resident EPG state: F+ , F- , Z  (complex, K=32) = 192 VGPRs.
    float fpx[KS], fpy[KS], fmx[KS], fmy[KS], zx[KS], zy[KS];
#pragma unroll
    for (int k = 0; k < KS; ++k) {
        fpx[k] = 0.f; fpy[k] = 0.f; fmx[k] = 0.f;
        fmy[k] = 0.f; zx[k]  = 0.f; zy[k]  = 0.f;
    }
    zx[0] = 1.0f;                       // initial Z0 = 1

    float* op = out + v * NP2;

#pragma unroll 1
    for (int p = 0; p < NP2; ++p) {
        // Wave-uniform rotation coefficients: LDS -> SGPRs.
        const float* a = &As[p * ASTR];
        const float ca2  = rfl(a[0]),  cca  = rfl(a[1]);
        const float r01r = rfl(a[2]),  r01i = rfl(a[3]);
        const float r02r = rfl(a[4]),  r02i = rfl(a[5]);
        const float r10r = rfl(a[6]),  r10i = rfl(a[7]);
        const float r12r = rfl(a[8]),  r12i = rfl(a[9]);
        const float r20r = rfl(a[10]), r20i = rfl(a[11]);
        const float r21r = rfl(a[12]), r21i = rfl(a[13]);

        // Active band: only k <= min(p, 63-p, 31) can influence any F0 output
        // (dependency cone moves <=1 k per pulse). Output-exact skipping.
        int ka = min(p, (NP2 - 1) - p);
        ka = min(ka, KS - 1);

        // ---- RF rotation (block-diag 3x3 complex), in place. F0 = P at k=0.
        // Z relaxation (and recovery at k=0) folded in: Z never shifts.
        float f0x = 0.f, f0y = 0.f;
#pragma unroll
        for (int k = 0; k < KS; ++k) {
            if (k <= ka) {   // uniform branch (p is wave-uniform)
                const float ax = fpx[k], ay = fpy[k];
                const float bx = fmx[k], by = fmy[k];
                const float cx = zx[k],  cy = zy[k];
                const float Px = ca2*ax + r01r*bx - r01i*by + r02r*cx - r02i*cy;
                const float Py = ca2*ay + r01r*by + r01i*bx + r02r*cy + r02i*cx;
                const float Mx = r10r*ax - r10i*ay + ca2*bx + r12r*cx - r12i*cy;
                const float My = r10r*ay + r10i*ax + ca2*by + r12r*cy + r12i*cx;
                const float Qx = r20r*ax - r20i*ay + r21r*bx - r21i*by + cca*cx;
                const float Qy = r20r*ay + r20i*ax + r21r*by + r21i*bx + cca*cy;
                fpx[k] = Px; fpy[k] = Py;
                fmx[k] = Mx; fmy[k] = My;
                if (k == 0) {
                    f0x = Px; f0y = Py;
                    zx[0] = E1 * Qx + B0;   // recovery on Z0 (real)
                    zy[0] = E1 * Qy;
                } else {
                    zx[k] = E1 * Qx;
                    zy[k] = E1 * Qy;
                }
            }
        }

        op[p] = PD * sqrtf(f0x * f0x + f0y * f0y);

        // ---- Shift + relaxation of F states, in place.
        const float tx = fmx[1], ty = fmy[1];          // rotated F-_1 (old if ka==0)
#pragma unroll
        for (int k = 0; k < KS - 1; ++k)               // F-_k <- E2 * F-_{k+1}
            if (k <= ka) { fmx[k] = E2 * fmx[k + 1]; fmy[k] = E2 * fmy[k + 1]; }
        if (ka == KS - 1) { fmx[KS - 1] = 0.f; fmy[KS - 1] = 0.f; }
#pragma unroll
        for (int k = KS - 1; k >= 1; --k)              // F+_k <- E2 * F+_{k-1}
            if (k <= ka + 1) { fpx[k] = E2 * fpx[k - 1]; fpy[k] = E2 * fpy[k - 1]; }
        fpx[0] = E2 * tx;                              // F+_0 = conj(E2 * F-_1)
        fpy[0] = -E2 * ty;
    }
}

// ---------------------------------------------------------------------------
// Precompute the 64 RF rotation matrices from theta.
// Table layout per pulse (16 floats, 64B aligned rows):
//   [0]=ca2 (r00=r11, real) [1]=ca (r22, real)
//   [2,3]=r01   [4,5]=r02   [6,7]=r10   [8,9]=r12   [10,11]=r20  [12,13]=r21
// ---------------------------------------------------------------------------
__global__ __launch_bounds__(64) void epg_rot_precompute(
    const float* __restrict__ thr, const float* __restrict__ thi,
    float* __restrict__ Atab)
{
    int p = threadIdx.x;
    if (p >= NP2) return;
    float tr = thr[p], ti = thi[p];
    float a   = sqrtf(tr * tr + ti * ti);     // flip angle
    float phi = atan2f(ti, tr);               // phase
    float sa, ca;  sincosf(a, &sa, &ca);
    float sh, ch;  sincosf(0.5f * a, &sh, &ch);
    float ca2 = ch * ch;
    float sa2 = sh * sh;
    float s1, c1;  sincosf(phi, &s1, &c1);
    float c2 = c1 * c1 - s1 * s1;             // cos(2 phi)
    float s2 = 2.0f * s1 * c1;                // sin(2 phi)

    float* A = Atab + p * ASTR;
    A[0]  = ca2;              A[1]  = ca;
    A[2]  = sa2 * c2;         A[3]  = sa2 * s2;          // r01
    A[4]  = sa * s1;          A[5]  = -sa * c1;          // r02
    A[6]  = sa2 * c2;         A[7]  = -sa2 * s2;         // r10
    A[8]  = sa * s1;          A[9]  = sa * c1;           // r12
    A[10] = -0.5f * sa * s1;  A[11] = -0.5f * sa * c1;   // r20
    A[12] = -0.5f * sa * s1;  A[13] =  0.5f * sa * c1;   // r21
    A[14] = 0.0f;             A[15] = 0.0f;
}

// ---------------------------------------------------------------------------
extern "C" void kernel_launch(void* const* d_in, const int* in_sizes, int n_in,
                              void* d_out, int out_size, void* d_ws, size_t ws_size,
                              hipStream_t stream) {
    const float* theta_re = (const float*)d_in[0];   // [64]
    const float* theta_im = (const float*)d_in[1];   // [64]
    const float* TR       = (const float*)d_in[2];   // [1]
    const float* qmaps    = (const float*)d_in[3];   // [3*32768]
    float* out  = (float*)d_out;                     // [32768*64]
    float* Atab = (float*)d_ws;                      // 64*16 floats = 4 KB

    hipLaunchKernelGGL(epg_rot_precompute, dim3(1), dim3(64), 0, stream,
                       theta_re, theta_im, Atab);
    hipLaunchKernelGGL(epg_main, dim3(NV / 256), dim3(256), 0, stream,
                       Atab, qmaps, TR, out);
}